// RUNG_homophily_adaptive_21002390078169
// MI455X (gfx1250) — compile-verified
//
#include <hip/hip_runtime.h>
#include <hip/hip_bf16.h>

#define NN    4096
#define INF   512
#define HID   64
#define OUTC  16
#define MAXD  256
#define EPSF  1e-8f
#define LAMV  (1.0f/0.9f - 1.0f)   /* 1/lam_hat - 1 */
#define ASCAD 3.7f

typedef __attribute__((ext_vector_type(2))) float v2f;
typedef __attribute__((ext_vector_type(4))) float v4f;
typedef __attribute__((ext_vector_type(8))) float v8f;

// ---------------------------------------------------------------------------
// SCAD MM weight  w = rho'(y)/y  with per-edge lambda (reference _scad_weight)
// ---------------------------------------------------------------------------
__device__ __forceinline__ float scad_w(float y, float lam) {
    float ys = fmaxf(y, EPSF);
    float w;
    if (y <= lam)              w = lam / ys;
    else if (y <= ASCAD * lam) w = (ASCAD * lam - y) / ((ASCAD - 1.0f) * ys);
    else                       w = 0.0f;
    if (w != w) w = 1.0f;      // NaN -> 1 (reference)
    return w;
}

// ---------------------------------------------------------------------------
// Fused MLP: F0 = ReLU(X@W1 + b1)@W2 + b2     using V_WMMA_F32_16X16X4_F32
// block = 128 threads (4 waves), computes 16 rows. Wave w does cols [16w,16w+16)
// of the 16x64 hidden tile; wave 0 then does the 16x16 second GEMM from LDS.
// ---------------------------------------------------------------------------
__global__ __launch_bounds__(128)
void k_mlp(const float* __restrict__ X,  const float* __restrict__ W1,
           const float* __restrict__ b1, const float* __restrict__ W2,
           const float* __restrict__ b2, float* __restrict__ F0) {
    __shared__ float h1[16][HID + 1];   // +1 pad: avoid 16-way bank conflicts
    const int wave = threadIdx.x >> 5;
    const int lid  = threadIdx.x & 31;
    const int m    = lid & 15;          // A-matrix row (M) / C col (N)
    const int kh   = (lid >> 4) << 1;   // K half-select: 0 or 2
    const int row0 = blockIdx.x * 16;
    const int ncol = wave * 16 + (lid & 15);

    v8f c = {};
    for (int k = 0; k < INF; k += 4) {
        v2f a, b;
        a.x = X[(size_t)(row0 + m) * INF + k + kh];
        a.y = X[(size_t)(row0 + m) * INF + k + kh + 1];
        b.x = W1[(size_t)(k + kh) * HID + ncol];
        b.y = W1[(size_t)(k + kh + 1) * HID + ncol];
        c = __builtin_amdgcn_wmma_f32_16x16x4_f32(false, a, false, b,
                                                  (short)0, c, false, false);
    }
    const float bias = b1[ncol];
    const int mbase = (lid >> 4) * 8;   // C layout: lanes>=16 hold M=8..15
#pragma unroll
    for (int r = 0; r < 8; ++r) {
        float v = c[r] + bias;
        h1[mbase + r][ncol] = v > 0.0f ? v : 0.0f;
    }
    __syncthreads();

    if (wave == 0) {                    // whole wave -> EXEC all ones for WMMA
        v8f c2 = {};
        for (int k = 0; k < HID; k += 4) {
            v2f a, b;
            a.x = h1[m][k + kh];
            a.y = h1[m][k + kh + 1];
            b.x = W2[(k + kh) * OUTC + (lid & 15)];
            b.y = W2[(k + kh + 1) * OUTC + (lid & 15)];
            c2 = __builtin_amdgcn_wmma_f32_16x16x4_f32(false, a, false, b,
                                                       (short)0, c2, false, false);
        }
        const float bias2 = b2[lid & 15];
#pragma unroll
        for (int r = 0; r < 8; ++r)
            F0[(size_t)(row0 + mbase + r) * OUTC + (lid & 15)] = c2[r] + bias2;
    }
}

// ---------------------------------------------------------------------------
// Edge-list build: one block per row, stream the 64MB adjacency exactly once.
// b128 non-temporal loads: 512B per wave-issue, no L2 pollution (A read once).
// deg = #off-diagonal nonzeros; D = deg + 1 (forced self-loop).
// ---------------------------------------------------------------------------
__global__ __launch_bounds__(256)
void k_edges(const float* __restrict__ A, int* __restrict__ cols,
             int* __restrict__ deg, float* __restrict__ Dval,
             float* __restrict__ rsD) {
    __shared__ int cnt;
    const int i = blockIdx.x;
    if (threadIdx.x == 0) cnt = 0;
    __syncthreads();
    const v4f* Arow = (const v4f*)(A + (size_t)i * NN);   // row base 16KB-aligned
#pragma unroll
    for (int q = threadIdx.x; q < NN / 4; q += 256) {
        v4f a = __builtin_nontemporal_load(&Arow[q]);
        const int j0 = q * 4;
#pragma unroll
        for (int t = 0; t < 4; ++t) {
            int j = j0 + t;
            if (a[t] > 0.0f && j != i) {
                int s = atomicAdd(&cnt, 1);
                if (s < MAXD) cols[i * MAXD + s] = j;
            }
        }
    }
    __syncthreads();
    if (threadIdx.x == 0) {
        int d = cnt > MAXD ? MAXD : cnt;
        deg[i] = d;
        float D = (float)d + 1.0f;
        Dval[i] = D;
        rsD[i]  = rsqrtf(D);
    }
}

// ---------------------------------------------------------------------------
// Row softmax of F0 (16-wide) -> P
// ---------------------------------------------------------------------------
__global__ __launch_bounds__(256)
void k_softmax(const float* __restrict__ F0, float* __restrict__ P) {
    int i = blockIdx.x * blockDim.x + threadIdx.x;
    if (i >= NN) return;
    float v[OUTC], mx = -3.4e38f;
#pragma unroll
    for (int c = 0; c < OUTC; ++c) { v[c] = F0[i * OUTC + c]; mx = fmaxf(mx, v[c]); }
    float s = 0.0f;
#pragma unroll
    for (int c = 0; c < OUTC; ++c) { v[c] = __expf(v[c] - mx); s += v[c]; }
    float inv = 1.0f / s;
#pragma unroll
    for (int c = 0; c < OUTC; ++c) P[i * OUTC + c] = v[c] * inv;
}

// ---------------------------------------------------------------------------
// Per-node homophily h -> adaptive quantile q -> sorted index idx (fixed).
// sim_i = sum_{j in edges(i)} <P_i, P_j>   (only edge entries of P@P^T matter)
// ---------------------------------------------------------------------------
__global__ __launch_bounds__(32)
void k_quant(const float* __restrict__ P, const int* __restrict__ cols,
             const int* __restrict__ deg, int* __restrict__ idxq) {
    const int i = blockIdx.x, l = threadIdx.x;
    const int d = deg[i];
    float pi[OUTC];
#pragma unroll
    for (int c = 0; c < OUTC; ++c) pi[c] = P[i * OUTC + c];
    float sim = 0.0f;
    for (int e = l; e < d; e += 32) {
        int j = cols[i * MAXD + e];
        float dot = 0.0f;
#pragma unroll
        for (int c = 0; c < OUTC; ++c) dot += pi[c] * P[j * OUTC + c];
        sim += dot;
    }
    for (int off = 16; off > 0; off >>= 1) sim += __shfl_down(sim, off, 32);
    if (l == 0) {
        const float minh = 1.0f / (float)OUTC;
        float h = minh;
        if (d > 0) h = fminf(fmaxf(sim / fmaxf((float)d, 1.0f), minh), 1.0f);
        float q = fminf(fmaxf(0.75f + (1.0f - h) * 0.2f, 0.75f), 0.99f);
        float degf = fmaxf((float)d, 1.0f);
        int idx = (int)floorf(q * (degf - 1.0f));
        idx = idx < 0 ? 0 : (idx > NN - 1 ? NN - 1 : idx);
        idxq[i] = idx;
    }
}

// ---------------------------------------------------------------------------
// F_unit_i = (F_i / sqrt(D_i)) / max(||F_i/sqrt(D_i)||, EPS)
// ---------------------------------------------------------------------------
__global__ __launch_bounds__(256)
void k_funit(const float* __restrict__ F, const float* __restrict__ rsD,
             float* __restrict__ U) {
    int i = blockIdx.x * blockDim.x + threadIdx.x;
    if (i >= NN) return;
    float rs = rsD[i], fn[OUTC], ss = 0.0f;
#pragma unroll
    for (int c = 0; c < OUTC; ++c) { fn[c] = F[i * OUTC + c] * rs; ss += fn[c] * fn[c]; }
    float inv = 1.0f / fmaxf(sqrtf(ss), EPSF);
#pragma unroll
    for (int c = 0; c < OUTC; ++c) U[i * OUTC + c] = fn[c] * inv;
}

// ---------------------------------------------------------------------------
// Per row: edge cosine distances y (stored), then the idx-th smallest edge y
// via in-wave rank-by-counting selection -> lam_i = max(gamma,EPS)/a.
// ---------------------------------------------------------------------------
__global__ __launch_bounds__(32)
void k_gamma(const float* __restrict__ U, const int* __restrict__ cols,
             const int* __restrict__ deg, const int* __restrict__ idxq,
             float* __restrict__ ey, float* __restrict__ lam) {
    __shared__ float yl[MAXD];
    __shared__ float gsh;
    const int i = blockIdx.x, l = threadIdx.x;
    const int d = deg[i];
    float ui[OUTC];
#pragma unroll
    for (int c = 0; c < OUTC; ++c) ui[c] = U[i * OUTC + c];
    for (int e = l; e < d; e += 32) {
        int j = cols[i * MAXD + e];
        float dot = 0.0f;
#pragma unroll
        for (int c = 0; c < OUTC; ++c) dot += ui[c] * U[j * OUTC + c];
        float y = fminf(fmaxf(1.0f - dot, 0.0f), 2.0f);
        ey[i * MAXD + e] = y;
        yl[e] = y;
    }
    if (l == 0) gsh = EPSF;             // deg==0 rows: gamma irrelevant to output
    __syncthreads();
    const int target = idxq[i];
    for (int e = l; e < d; e += 32) {
        float ye = yl[e];
        int rank = 0;
        for (int k = 0; k < d; ++k) {
            float yk = yl[k];
            rank += (yk < ye) || (yk == ye && k < e);   // unique ranks on ties
        }
        if (rank == target) gsh = ye;   // exactly one lane matches
    }
    __syncthreads();
    if (l == 0) lam[i] = fmaxf(gsh, EPSF) / ASCAD;
}

// ---------------------------------------------------------------------------
// Row update:  Q_i = (sum_e w_e)/D_i + LAM
//   F'_i = (sum_e w_e * rsD_i * rsD_j * F_j)/Q_i + LAM*F0_i/Q_i
// 16 lanes = 16 output components; edges split across the two half-waves.
// ---------------------------------------------------------------------------
__global__ __launch_bounds__(32)
void k_update(const float* __restrict__ F,   const float* __restrict__ F0,
              const float* __restrict__ ey,  const int* __restrict__ cols,
              const int* __restrict__ deg,   const float* __restrict__ lam,
              const float* __restrict__ rsD, const float* __restrict__ Dval,
              float* __restrict__ Fn) {
    const int i = blockIdx.x, l = threadIdx.x;
    const int c = l & 15, half = l >> 4;
    const int d = deg[i];
    const float lami = lam[i];
    float acc = 0.0f, wsum = 0.0f;
    for (int e = half; e < d; e += 2) {
        int j   = cols[i * MAXD + e];
        float y = ey[i * MAXD + e];
        float w = scad_w(y, fmaxf(lami, lam[j]));
        if (c == 0) wsum += w;
        acc += w * rsD[j] * F[j * OUTC + c];
    }
    acc  += __shfl_down(acc, 16, 32);   // fold half-waves: lanes 0-15 complete
    wsum += __shfl_down(wsum, 16, 32);
    wsum  = __shfl(wsum, 0, 32);
    const float Q = wsum / Dval[i] + LAMV;
    if (l < 16)
        Fn[i * OUTC + c] = (acc * rsD[i]) / Q + LAMV * F0[i * OUTC + c] / Q;
}

// ---------------------------------------------------------------------------
extern "C" void kernel_launch(void* const* d_in, const int* in_sizes, int n_in,
                              void* d_out, int out_size, void* d_ws, size_t ws_size,
                              hipStream_t stream) {
    const float* A  = (const float*)d_in[0];
    const float* X  = (const float*)d_in[1];
    const float* W1 = (const float*)d_in[2];
    const float* b1 = (const float*)d_in[3];
    const float* W2 = (const float*)d_in[4];
    const float* b2 = (const float*)d_in[5];

    // workspace layout (~9.5 MB)
    float* F0   = (float*)d_ws;
    float* bufA = F0   + (size_t)NN * OUTC;
    float* U    = bufA + (size_t)NN * OUTC;
    float* P    = U    + (size_t)NN * OUTC;
    float* lam  = P    + (size_t)NN * OUTC;
    float* Dval = lam  + NN;
    float* rsD  = Dval + NN;
    int*   deg  = (int*)(rsD + NN);
    int*   idxq = deg  + NN;
    int*   cols = idxq + NN;
    float* ey   = (float*)(cols + (size_t)NN * MAXD);

    k_mlp    <<<NN / 16, 128, 0, stream>>>(X, W1, b1, W2, b2, F0);
    k_edges  <<<NN, 256, 0, stream>>>(A, cols, deg, Dval, rsD);
    k_softmax<<<NN / 256, 256, 0, stream>>>(F0, P);
    k_quant  <<<NN, 32, 0, stream>>>(P, cols, deg, idxq);

    const float* cur = F0;
    for (int s = 0; s < 10; ++s) {
        float* nxt = (s & 1) ? (float*)d_out : bufA;   // step 10 (s=9) -> d_out
        k_funit <<<NN / 256, 256, 0, stream>>>(cur, rsD, U);
        k_gamma <<<NN, 32, 0, stream>>>(U, cols, deg, idxq, ey, lam);
        k_update<<<NN, 32, 0, stream>>>(cur, F0, ey, cols, deg, lam, rsD, Dval, nxt);
        cur = nxt;
    }
    (void)in_sizes; (void)n_in; (void)out_size; (void)ws_size;
}